// CNFBlock_25967372272227
// MI455X (gfx1250) — compile-verified
//
#include <hip/hip_runtime.h>
#include <math.h>
#include <stdint.h>

// ---------------------------------------------------------------------------
// CNF log-prob head for MI455X (gfx1250), wave32 + V_WMMA_F32_16X16X4_F32,
// with TDM (tensor_load_to_lds) staging of the 16x256 fp32 tiles.
// out[i,t] = dot(h_i, z_t) - 0.5*||z_t||^2 - 0.5*||h_i||^2 - 128*log(2pi) - delta_t
// delta_t  = -0.5*(tr0_t + tr1_t) from 2-step Euler CNF trace estimate.
// ---------------------------------------------------------------------------

typedef __attribute__((ext_vector_type(2))) float v2f;
typedef __attribute__((ext_vector_type(4))) float v4f;
typedef __attribute__((ext_vector_type(8))) float v8f;

#define SB      1024          // SEQ*BATCH
#define DDIM    256
#define NTOKEN  50257
#define NT_TILES ((NTOKEN + 15) / 16)   // 3142
#define ZPAD    260           // 256 + 4 pad -> conflict-free ds_load_b64 across lanes
#define LOG_2PI 1.8378770664093453f
#define CONST_TERM (0.5f * (float)DDIM * LOG_2PI)   // 128*log(2pi)

__device__ __forceinline__ v8f wmma_f32(v2f a, v2f b, v8f c) {
  // 8 args: (neg_a, A, neg_b, B, c_mod, C, reuse_a, reuse_b)
  return __builtin_amdgcn_wmma_f32_16x16x4_f32(false, a, false, b, (short)0, c,
                                               false, false);
}

// ---------------------------------------------------------------------------
// TDM staging: DMA a 16x256 fp32 tile (rows of a row-major [*,256] tensor)
// into LDS, inserting 4 pad DWORDs after every 256 DWORDs so the LDS layout
// matches float[16][ZPAD].  OOB rows (>= rows_remaining) are zero-filled by
// the TDM, which is what the masked epilogues expect on the ragged last tile.
// ---------------------------------------------------------------------------
#if defined(__AMDGCN__) && __has_builtin(__builtin_amdgcn_tensor_load_to_lds)
#define USE_TDM 1
typedef __attribute__((ext_vector_type(4))) unsigned int u32x4;
typedef __attribute__((ext_vector_type(8))) int i32x8;
typedef __attribute__((ext_vector_type(4))) int i32x4;

__device__ __forceinline__ void tdm_stage_tile(void* lds_dst, const float* gsrc,
                                               int rows_remaining) {
  u32x4 g0 = {};
  i32x8 g1 = {};
  i32x4 g2 = {};   // groups 2/3 unused for 2D tensors
  i32x4 g3 = {};
  const unsigned long long ga = (unsigned long long)(uintptr_t)gsrc;
  // ---- D# group 0: count=1, lds_addr, global_addr[56:0], type=2 ----
  g0[0] = 1u;                                        // count=1, user mode
  g0[1] = (unsigned)(uintptr_t)lds_dst;              // lds_addr (bytes)
  g0[2] = (unsigned)ga;                              // global_addr[31:0]
  g0[3] = (unsigned)((ga >> 32) & 0x01FFFFFFu) | (2u << 30);  // [56:32] | type=2
  // ---- D# group 1 ----
  // data_size=2 (4B), pad_enable, pad_interval=7 (256 DW), pad_amount=3 (4 DW)
  g1[0] = (int)((2u << 16) | (1u << 20) | (7u << 22) | (3u << 25));
  g1[1] = (int)(256u << 16);                         // tensor_dim0=256 (bits 79:48)
  g1[2] = (int)(((unsigned)rows_remaining & 0xFFFFu) << 16);   // tensor_dim1 lo
  g1[3] = (int)((((unsigned)rows_remaining >> 16) & 0xFFFFu)   // tensor_dim1 hi
                | (256u << 16));                     // tile_dim0=256
  g1[4] = 16;                                        // tile_dim1=16
  g1[5] = 256;                                       // tensor_dim0_stride=256
#if __clang_major__ >= 23
  i32x8 g4 = {};
  __builtin_amdgcn_tensor_load_to_lds(g0, g1, g2, g3, g4, 0);
#else
  __builtin_amdgcn_tensor_load_to_lds(g0, g1, g2, g3, 0);
#endif
  __builtin_amdgcn_s_wait_tensorcnt(0);
}
#endif

// ---------------------------------------------------------------------------
// Tiny aux kernels: row norms of h, and diag(W1x @ W2).
// ---------------------------------------------------------------------------
__global__ void hnorm_kernel(const float* __restrict__ h, float* __restrict__ hnorm) {
  int r = blockIdx.x * blockDim.x + threadIdx.x;
  if (r >= SB) return;
  const v4f* p = (const v4f*)(h + (size_t)r * DDIM);
  float s = 0.f;
#pragma unroll 4
  for (int i = 0; i < DDIM / 4; ++i) {
    v4f v = p[i];
    s += v.x * v.x + v.y * v.y + v.z * v.z + v.w * v.w;
  }
  hnorm[r] = s;
}

// diagM[j] = sum_i W1x[j,i] * W2[i,j]
__global__ void diagm_kernel(const float* __restrict__ W1x,
                             const float* __restrict__ W2,
                             float* __restrict__ diagM) {
  int j = blockIdx.x * blockDim.x + threadIdx.x;
  if (j >= DDIM) return;
  float s = 0.f;
  for (int i = 0; i < DDIM; ++i)
    s = fmaf(W1x[(size_t)j * DDIM + i], W2[(size_t)i * DDIM + j], s);
  diagM[j] = s;
}

// ---------------------------------------------------------------------------
// CNF delta kernel: one block per 16-token tile, 8 wave32's.
// All GEMMs (16 x 256 x 256) run on V_WMMA_F32_16X16X4_F32.
// Also emits znorm (uses original z0).
// ---------------------------------------------------------------------------
__global__ __launch_bounds__(256)
void cnf_delta_kernel(const float* __restrict__ emb,
                      const float* __restrict__ W1x,
                      const float* __restrict__ w1t,
                      const float* __restrict__ b1,
                      const float* __restrict__ W2,
                      const float* __restrict__ b2,
                      const float* __restrict__ diagM,
                      float* __restrict__ delta,
                      float* __restrict__ znorm_out) {
  __shared__ float zs [16][ZPAD];   // z  (updated in-place between Euler steps)
  __shared__ float sps[16][ZPAD];   // softplus(pre) staging
  __shared__ float tr [16];         // trace accumulators (tr0+tr1)
  __shared__ float part[256];       // ||z||^2 partials
  __shared__ float zn [16];

  const int tid = threadIdx.x;
  const int t0  = blockIdx.x * 16;

  // ---- stage 16 token rows + squared-norm partials -------------------------
#ifdef USE_TDM
  if (tid < 32) tdm_stage_tile(&zs[0][0], emb + (size_t)t0 * DDIM, NTOKEN - t0);
  __syncthreads();
  {
    const int row = tid >> 4;
    const int cb  = (tid & 15) * 16;
    float s = 0.f;
#pragma unroll
    for (int i = 0; i < 16; ++i) { const float v = zs[row][cb + i]; s = fmaf(v, v, s); }
    part[tid] = s;
  }
#else
  {
    int row = tid >> 4;
    int cb  = (tid & 15) * 16;
    int tok = t0 + row;
    if (tok >= NTOKEN) tok = NTOKEN - 1;           // clamp (values masked later)
    const float* src = emb + (size_t)tok * DDIM + cb;
    float s = 0.f;
#pragma unroll
    for (int i = 0; i < 16; i += 4) {
      v4f v = *(const v4f*)(src + i);
      zs[row][cb + i + 0] = v.x;
      zs[row][cb + i + 1] = v.y;
      zs[row][cb + i + 2] = v.z;
      zs[row][cb + i + 3] = v.w;
      s += v.x * v.x + v.y * v.y + v.z * v.z + v.w * v.w;
    }
    part[tid] = s;
  }
#endif
  if (tid < 16) tr[tid] = 0.f;
  __syncthreads();
  if (tid < 16) {
    float s = 0.f;
#pragma unroll
    for (int j = 0; j < 16; ++j) s += part[tid * 16 + j];
    zn[tid] = s;
  }

  const int lane = tid & 31;
  const int wave = tid >> 5;       // 0..7
  const int half = lane >> 4;      // lane 16..31 handles K+2 / M+8
  const int lm   = lane & 15;

  // this wave owns N-tiles {2*wave, 2*wave+1}; per-lane column constants
  const int c0 = wave * 32 + lm;
  const int c1 = c0 + 16;
  const float dg0 = diagM[c0], dg1 = diagM[c1];
  const float b10 = b1[c0],    b11 = b1[c1];
  const float wt0 = w1t[c0],   wt1 = w1t[c1];
  const float b20 = b2[c0],    b21 = b2[c1];

  // 16x16x256 WMMA accumulate: A from LDS (rows=tokens), B[k][n]=W[col][k]
  auto gemm = [&](float (*A)[ZPAD], const float* __restrict__ W, int col) -> v8f {
    v8f acc = {};
    const float* arow = &A[lm][2 * half];
    const float* brow = W + (size_t)col * DDIM + 2 * half;
#pragma unroll 8
    for (int k = 0; k < DDIM; k += 4) {
      v2f a; a.x = arow[k]; a.y = arow[k + 1];      // ds_load
      v2f b = *(const v2f*)(brow + k);              // global b64 (L2-resident W)
      acc = wmma_f32(a, b, acc);
    }
    return acc;
  };

  // pre = z @ W1x^T + t*w1t + b1 ; trace += sigmoid(pre).diagM ; opt. store sp
  auto stepA = [&](float tcoef, bool store_sp) {
    float trp[8];
#pragma unroll
    for (int v = 0; v < 8; ++v) trp[v] = 0.f;
#pragma unroll
    for (int tcol = 0; tcol < 2; ++tcol) {
      const int   c    = tcol ? c1 : c0;
      const float bias = (tcol ? b11 : b10) + tcoef * (tcol ? wt1 : wt0);
      const float dg   = tcol ? dg1 : dg0;
      v8f acc = gemm(zs, W1x, c);
#pragma unroll
      for (int v = 0; v < 8; ++v) {
        const int   m   = v + 8 * half;
        const float pre = acc[v] + bias;
        const float e   = __expf(-fabsf(pre));
        const float sig = (pre >= 0.f) ? 1.f / (1.f + e) : e / (1.f + e);
        trp[v] += sig * dg;
        if (store_sp) sps[m][c] = fmaxf(pre, 0.f) + log1pf(e);   // stable softplus
      }
    }
#pragma unroll
    for (int v = 0; v < 8; ++v) atomicAdd(&tr[v + 8 * half], trp[v]);  // ds_add_f32
  };

  // f = softplus(pre) @ W2^T + b2 ; z += 0.5*f   (each (m,c) owned by one lane)
  auto stepB = [&]() {
#pragma unroll
    for (int tcol = 0; tcol < 2; ++tcol) {
      const int   c    = tcol ? c1 : c0;
      const float bias = tcol ? b21 : b20;
      v8f acc = gemm(sps, W2, c);
#pragma unroll
      for (int v = 0; v < 8; ++v) {
        const int m = v + 8 * half;
        zs[m][c] += 0.5f * (acc[v] + bias);
      }
    }
  };

  stepA(0.0f, true);     // Euler step 0: t=0, keep softplus for f0
  __syncthreads();
  stepB();               // z1 = z0 + 0.5*f0
  __syncthreads();
  stepA(0.5f, false);    // Euler step 1: t=0.5, trace only
  __syncthreads();

  if (tid < 16) {
    const int tok = t0 + tid;
    if (tok < NTOKEN) {
      delta[tok]     = -0.5f * tr[tid];   // delta = -dt*(tr0+tr1), dt=0.5
      znorm_out[tok] = zn[tid];
    }
  }
}

// ---------------------------------------------------------------------------
// Main kernel: out(1024 x 50257) = h @ emb^T with fused MVN/CNF epilogue.
// Grid: (3142 N-tiles, 8 M-blocks of 128 rows); 8 wave-tiles (16x16) per WG.
// emb tile staged into LDS by the TDM and shared by the 8 waves; h is served
// from L0/L2 (1 MB, fully resident).
// ---------------------------------------------------------------------------
__global__ __launch_bounds__(256)
void logp_kernel(const float* __restrict__ h,
                 const float* __restrict__ emb,
                 const float* __restrict__ hnorm,
                 const float* __restrict__ znorm,
                 const float* __restrict__ delta,
                 float* __restrict__ out) {
  __shared__ float bs[16][ZPAD];

  const int tid = threadIdx.x;
  const int n0  = blockIdx.x * 16;

  // ---- stage 16 emb rows (the B tile, shared by all 8 waves) ---------------
#ifdef USE_TDM
  if (tid < 32) tdm_stage_tile(&bs[0][0], emb + (size_t)n0 * DDIM, NTOKEN - n0);
  __syncthreads();
#else
  {
    int row = tid >> 4;
    int cb  = (tid & 15) * 16;
    int tok = n0 + row;
    if (tok >= NTOKEN) tok = NTOKEN - 1;           // clamp, stores masked below
    const float* src = emb + (size_t)tok * DDIM + cb;
#pragma unroll
    for (int i = 0; i < 16; i += 4) {
      v4f v = *(const v4f*)(src + i);
      bs[row][cb + i + 0] = v.x;
      bs[row][cb + i + 1] = v.y;
      bs[row][cb + i + 2] = v.z;
      bs[row][cb + i + 3] = v.w;
    }
  }
  __syncthreads();
#endif

  const int lane = tid & 31;
  const int wave = tid >> 5;
  const int half = lane >> 4;
  const int lm   = lane & 15;
  const int m0   = blockIdx.y * 128 + wave * 16;

  const float* arow = h + (size_t)(m0 + lm) * DDIM + 2 * half;
  const float* brow = &bs[lm][2 * half];

  v8f acc = {};
#pragma unroll 8
  for (int k = 0; k < DDIM; k += 4) {
    v2f a = *(const v2f*)(arow + k);   // global b64, h is L0/L2 resident (1MB)
    v2f b; b.x = brow[k]; b.y = brow[k + 1];   // ds_load from staged tile
    acc = wmma_f32(a, b, acc);
  }

  const int t  = n0 + lm;
  const int tc = (t < NTOKEN) ? t : (NTOKEN - 1);
  const float colterm = -0.5f * znorm[tc] - delta[tc] - CONST_TERM;

  if (t < NTOKEN) {
#pragma unroll
    for (int v = 0; v < 8; ++v) {
      const int i = m0 + v + 8 * half;           // output row
      out[(size_t)i * NTOKEN + t] = acc[v] - 0.5f * hnorm[i] + colterm;
    }
  }
}

// ---------------------------------------------------------------------------
extern "C" void kernel_launch(void* const* d_in, const int* in_sizes, int n_in,
                              void* d_out, int out_size, void* d_ws, size_t ws_size,
                              hipStream_t stream) {
  const float* h   = (const float*)d_in[0];   // (32,32,256)
  const float* emb = (const float*)d_in[1];   // (50257,256)
  const float* W1x = (const float*)d_in[2];   // (256,256)
  const float* w1t = (const float*)d_in[3];   // (256,)
  const float* b1  = (const float*)d_in[4];   // (256,)
  const float* W2  = (const float*)d_in[5];   // (256,256)
  const float* b2  = (const float*)d_in[6];   // (256,)
  float* out = (float*)d_out;                 // (1024, 50257)

  float* ws    = (float*)d_ws;
  float* znorm = ws;                          // NTOKEN
  float* delta = ws + NTOKEN;                 // NTOKEN
  float* hnorm = ws + 2 * NTOKEN;             // SB
  float* diagM = ws + 2 * NTOKEN + SB;        // DDIM

  hipLaunchKernelGGL(hnorm_kernel, dim3(SB / 256), dim3(256), 0, stream, h, hnorm);
  hipLaunchKernelGGL(diagm_kernel, dim3(1), dim3(256), 0, stream, W1x, W2, diagM);
  hipLaunchKernelGGL(cnf_delta_kernel, dim3(NT_TILES), dim3(256), 0, stream,
                     emb, W1x, w1t, b1, W2, b2, diagM, delta, znorm);
  hipLaunchKernelGGL(logp_kernel, dim3(NT_TILES, SB / 128), dim3(256), 0, stream,
                     h, emb, hnorm, znorm, delta, out);
}